// SingleShotInhibition_22359599743467
// MI455X (gfx1250) — compile-verified
//
#include <hip/hip_runtime.h>

typedef float v2f __attribute__((ext_vector_type(2)));
typedef float v8f __attribute__((ext_vector_type(8)));

#define CCH 512               // channels (power of two -> & (CCH-1) for wrap)
#define HW  784               // 28*28 pixels per channel plane
#define NTAPS 27
#define KTILES 12             // K = 48 = 16-channel tile + 16-halo each side
#define WAVES_PER_BLOCK 4
#define PTILES_TOTAL 49       // 784 / 16
#define PTILES_PER_GROUP 4
#define NGROUPS 13            // ceil(49 / 4)

__global__ __launch_bounds__(128) void ssi_wmma_f32_kernel(
    const float* __restrict__ act,   // [B, 512, 28, 28]
    const float* __restrict__ filt,  // [27]
    float* __restrict__ out)         // [B, 512, 28, 28]
{
    __shared__ float sfilt[32];
    const int tid = threadIdx.x;
    if (tid < NTAPS) sfilt[tid] = filt[tid];
    __syncthreads();

    const int lane  = tid & 31;
    const int wave  = tid >> 5;
    const int m     = lane & 15;   // A-matrix row / B-N column / pixel index
    const int khalf = lane >> 4;   // 0 -> K pair {0,1}; 1 -> K pair {2,3}

    const int f0 = blockIdx.y << 4;      // output-channel tile base
    const int b  = blockIdx.z;           // batch

    const int group = blockIdx.x * WAVES_PER_BLOCK + wave;
    if (group >= NGROUPS) return;        // wave-uniform -> EXEC stays all-1s

    // ---- Build banded circular-Toeplitz A tiles (16x48 as 12 K-steps of 4).
    // A layout (ISA 7.12.2, 32-bit A 16x4): VGPR0 = K {0 | 2}, VGPR1 = K {1 | 3}
    // split by lane half; row M = lane % 16.
    v2f A[KTILES];
#pragma unroll
    for (int kk = 0; kk < KTILES; ++kk) {
#pragma unroll
        for (int j = 0; j < 2; ++j) {
            const int k = kk * 4 + khalf * 2 + j;            // local K in [0,48)
            const int c_abs = (f0 - 16 + k) & (CCH - 1);     // absolute in-channel
            const int f_abs = f0 + m;                        // absolute out-channel
            const int d = ((c_abs - f_abs + CCH / 2) & (CCH - 1)) - CCH / 2;
            float w = 0.0f;
            if (d != 0 && d >= -13 && d <= 13) w = sfilt[d + 13];
            A[kk][j] = w;
        }
    }

    const float* actb = act + (size_t)b * CCH * HW;
    float*       outb = out + (size_t)b * CCH * HW;

    int t0 = group * PTILES_PER_GROUP;
    int t1 = t0 + PTILES_PER_GROUP;
    if (t1 > PTILES_TOTAL) t1 = PTILES_TOTAL;

    // Band span is [f0-16, f0+31]; only f0==0 and f0==496 wrap around the
    // 512-channel circle. Block-uniform condition -> EXEC stays all-1s.
    const bool nowrap = (f0 >= 16) && (f0 <= CCH - 32);

    for (int t = t0; t < t1; ++t) {
        const int p = (t << 4) + m;        // pixel column (N index)

        // ---- C = residual: the activation tile itself (center tap is zero).
        // C/D layout: VGPR r holds M = r + 8*khalf, N = lane%16.
        const float* cbase = actb + (size_t)(f0 + khalf * 8) * HW + p;
        v8f acc;
#pragma unroll
        for (int r = 0; r < 8; ++r)
            acc[r] = cbase[(size_t)r * HW];

        // ---- 12 x V_WMMA_F32_16X16X4_F32 over the 48-channel band.
        // B layout (4x16): VGPR0 = K {0 | 2}, VGPR1 = K {1 | 3} by lane half,
        // column N = lane % 16.
        if (nowrap) {
            // Fast path: all B addresses are base + compile-time-constant
            // offsets -> folded into the 24-bit instruction IOFFSET.
            const float* bbase = actb + (size_t)(f0 - 16 + khalf * 2) * HW + p;
#pragma unroll
            for (int kk = 0; kk < KTILES; ++kk) {
                v2f Bv;
                Bv.x = bbase[(size_t)(kk * 4 + 0) * HW];
                Bv.y = bbase[(size_t)(kk * 4 + 1) * HW];
                acc = __builtin_amdgcn_wmma_f32_16x16x4_f32(
                    false, A[kk], false, Bv, (short)0, acc, false, false);
            }
        } else {
            // Wrap path (f-tiles 0 and 31 only): explicit circular indexing.
#pragma unroll
            for (int kk = 0; kk < KTILES; ++kk) {
                const int cb = f0 - 16 + kk * 4 + khalf * 2;
                const int c0 = cb & (CCH - 1);
                const int c1 = (cb + 1) & (CCH - 1);
                v2f Bv;
                Bv.x = actb[(size_t)c0 * HW + p];
                Bv.y = actb[(size_t)c1 * HW + p];
                acc = __builtin_amdgcn_wmma_f32_16x16x4_f32(
                    false, A[kk], false, Bv, (short)0, acc, false, false);
            }
        }

        // ---- Store D tile.
        float* obase = outb + (size_t)(f0 + khalf * 8) * HW + p;
#pragma unroll
        for (int r = 0; r < 8; ++r)
            obase[(size_t)r * HW] = acc[r];
    }
}

extern "C" void kernel_launch(void* const* d_in, const int* in_sizes, int n_in,
                              void* d_out, int out_size, void* d_ws, size_t ws_size,
                              hipStream_t stream) {
    const float* act  = (const float*)d_in[0];   // [B,512,28,28] fp32
    const float* filt = (const float*)d_in[1];   // [27] fp32
    float*       out  = (float*)d_out;           // [B,512,28,28] fp32

    const int B = in_sizes[0] / (CCH * HW);      // 64 for the reference shapes

    // grid.x * WAVES_PER_BLOCK must cover NGROUPS (13) wave-groups.
    dim3 grid((NGROUPS + WAVES_PER_BLOCK - 1) / WAVES_PER_BLOCK, CCH / 16, B);
    dim3 block(32 * WAVES_PER_BLOCK, 1, 1);      // 128 threads = 4 waves
    ssi_wmma_f32_kernel<<<grid, block, 0, stream>>>(act, filt, out);
}